// Transformer_3968549781875
// MI455X (gfx1250) — compile-verified
//
#include <hip/hip_runtime.h>
#include <hip/hip_bf16.h>

#define B_ 4
#define C_ 32
#define T_ 1024
#define U_ 32
#define H_ 128

typedef float v2f __attribute__((ext_vector_type(2)));
typedef float v8f __attribute__((ext_vector_type(8)));

// ---------------- helpers ----------------
__device__ __forceinline__ float wred_sum(float v) {
#pragma unroll
    for (int o = 16; o > 0; o >>= 1) v += __shfl_xor(v, o, 32);
    return v;
}
__device__ __forceinline__ float wred_max(float v) {
#pragma unroll
    for (int o = 16; o > 0; o >>= 1) v = fmaxf(v, __shfl_xor(v, o, 32));
    return v;
}
// tanh via single v_exp_f32; saturates correctly for |x| large (no inf/inf NaN)
__device__ __forceinline__ float fast_tanh(float x) {
    float e = __expf(2.0f * x);
    return 1.0f - 2.0f / (e + 1.0f);
}

// ---------------- kernel A: q = xt@Wt + bh (folded), k = xt@Wx ----------------
__global__ void qk_kernel(const float* __restrict__ x, const float* __restrict__ Wt,
                          const float* __restrict__ Wx, const float* __restrict__ bh,
                          float* __restrict__ q, float* __restrict__ k) {
    int idx = blockIdx.x * blockDim.x + threadIdx.x;   // (b*T+t)*U + u, u fastest -> Wt coalesced
    if (idx >= B_ * T_ * U_) return;
    int u  = idx & (U_ - 1);
    int bt = idx >> 5;                                  // U_ == 32
    int t  = bt & (T_ - 1);
    int b  = bt >> 10;                                  // T_ == 1024
    const float* xp = x + (size_t)b * C_ * T_ + t;      // x[b,c,t], stride T over c (wave-uniform value)
    float qa = 0.f, ka = 0.f;
#pragma unroll
    for (int c = 0; c < C_; ++c) {
        float xv = xp[(size_t)c * T_];
        qa = fmaf(xv, Wt[c * U_ + u], qa);
        ka = fmaf(xv, Wx[c * U_ + u], ka);
    }
    q[idx] = qa + bh[u];
    k[idx] = ka;
}

// ---------------- kernel B: scores + softmax -> a (the hot loop: 134M tanh) ----------------
__global__ void attn_kernel(const float* __restrict__ q, const float* __restrict__ k,
                            const float* __restrict__ Wa, const float* __restrict__ ba,
                            float* __restrict__ a_out) {
    const int row = blockIdx.x;          // b*T + i
    const int b   = row >> 10;
    const int tid = threadIdx.x;

    __shared__ float qs[U_];
    __shared__ float was[U_];
    __shared__ float red[8];

    if (tid < U_) { qs[tid] = q[row * U_ + tid]; was[tid] = Wa[tid]; }
    __syncthreads();

    const float bav = ba[0];
    const float* kb = k + (size_t)b * T_ * U_;

    float e[4];
    float m = -3.402823466e38f;
#pragma unroll
    for (int jj = 0; jj < 4; ++jj) {
        const int j = jj * 256 + tid;
        const float* kj = kb + (size_t)j * U_;
        if (jj < 3) __builtin_prefetch(kj + 256 * U_, 0, 0);   // global_prefetch_b8
        float acc = bav;
#pragma unroll
        for (int u = 0; u < U_; u += 4) {
            float4 kv = *(const float4*)(kj + u);
            acc = fmaf(was[u + 0], fast_tanh(qs[u + 0] + kv.x), acc);
            acc = fmaf(was[u + 1], fast_tanh(qs[u + 1] + kv.y), acc);
            acc = fmaf(was[u + 2], fast_tanh(qs[u + 2] + kv.z), acc);
            acc = fmaf(was[u + 3], fast_tanh(qs[u + 3] + kv.w), acc);
        }
        e[jj] = acc;
        m = fmaxf(m, acc);
    }

    // block max (8 waves)
    m = wred_max(m);
    if ((tid & 31) == 0) red[tid >> 5] = m;
    __syncthreads();
    float bm = red[0];
#pragma unroll
    for (int w = 1; w < 8; ++w) bm = fmaxf(bm, red[w]);

    float s = 0.f;
#pragma unroll
    for (int jj = 0; jj < 4; ++jj) { e[jj] = __expf(e[jj] - bm); s += e[jj]; }

    s = wred_sum(s);
    __syncthreads();
    if ((tid & 31) == 0) red[tid >> 5] = s;
    __syncthreads();
    float bs = 0.f;
#pragma unroll
    for (int w = 0; w < 8; ++w) bs += red[w];
    const float inv = 1.0f / (bs + 1e-5f);

    float* arow = a_out + (size_t)row * T_;
#pragma unroll
    for (int jj = 0; jj < 4; ++jj) arow[jj * 256 + tid] = e[jj] * inv;   // coalesced
}

// ---------------- kernel C: v = a @ xt via V_WMMA_F32_16X16X4_F32 ----------------
// one wave per 16x16 output tile; v stored in x-layout (B,C,T)
__global__ void av_wmma_kernel(const float* __restrict__ a, const float* __restrict__ x,
                               float* __restrict__ v) {
    const int tile = blockIdx.x;                 // b*(C/16)*(T/16) tiles = 512
    const int im   = tile & 63;                  // T/16 = 64
    const int cn   = (tile >> 6) & 1;            // C/16 = 2
    const int b    = tile >> 7;
    const int i0 = im * 16, c0 = cn * 16;

    const int lane = threadIdx.x & 31;
    const int half = lane >> 4;                  // 0: lanes 0-15, 1: lanes 16-31
    const int lm   = lane & 15;

    // A frag: lane row M = lm, holds K = 2*half, 2*half+1 (ISA 32-bit 16x4 layout)
    const float* arow = a + ((size_t)(b * T_) + i0 + lm) * T_ + 2 * half;
    // B frag: lane col N = lm, VGPR r holds K = r + 2*half  -> contiguous float2 in x[b, c0+lm, :]
    const float* xcol = x + ((size_t)b * C_ + c0 + lm) * T_ + 2 * half;

    v8f acc = {};
    for (int k0 = 0; k0 < T_; k0 += 4) {
        v2f af = *(const v2f*)(arow + k0);
        v2f bf = *(const v2f*)(xcol + k0);
        acc = __builtin_amdgcn_wmma_f32_16x16x4_f32(
            /*neg_a=*/false, af, /*neg_b=*/false, bf,
            /*c_mod=*/(short)0, acc, /*reuse_a=*/false, /*reuse_b=*/false);
    }

    // D: lane holds N = lm, VGPR r -> M = r + 8*half : 8 consecutive M -> two b128 stores
    float* vp = v + ((size_t)b * C_ + c0 + lm) * T_ + i0 + 8 * half;
    *(float4*)(vp + 0) = make_float4(acc[0], acc[1], acc[2], acc[3]);
    *(float4*)(vp + 4) = make_float4(acc[4], acc[5], acc[6], acc[7]);
}

// ---------------- kernel D: y = LN(x + v) over C=32 (one wave32 per (b,t)) ----------------
__global__ void ln1_kernel(const float* __restrict__ x, const float* __restrict__ v,
                           const float* __restrict__ gamma, const float* __restrict__ beta,
                           float* __restrict__ y) {
    const int w    = blockIdx.x * (blockDim.x >> 5) + (threadIdx.x >> 5);  // b*T + t
    const int lane = threadIdx.x & 31;                                      // channel c
    const int b = w >> 10, t = w & (T_ - 1);
    const size_t idx = ((size_t)b * C_ + lane) * T_ + t;
    float z = x[idx] + v[idx];
    float mean = wred_sum(z) * (1.0f / C_);
    float d = z - mean;
    float var = wred_sum(d * d) * (1.0f / C_) + 1e-14f;
    y[idx] = d * rsqrtf(var) * gamma[lane] + beta[lane];
}

// ---------------- kernel E: FF (relu GEMM + GEMM) fused with LN2, per (b,t) column ----------------
__global__ void ff_ln2_kernel(const float* __restrict__ y, const float* __restrict__ W1,
                              const float* __restrict__ b1, const float* __restrict__ W2,
                              const float* __restrict__ b2, const float* __restrict__ gamma2,
                              const float* __restrict__ beta2, float* __restrict__ y2) {
    __shared__ float ys[C_];
    __shared__ float h1s[H_];
    const int bt = blockIdx.x;
    const int b = bt >> 10, t = bt & (T_ - 1);
    const int tid = threadIdx.x;   // blockDim.x == H_ == 128

    if (tid < C_) ys[tid] = y[((size_t)b * C_ + tid) * T_ + t];
    __syncthreads();

    {   // h1[h] = relu( sum_c y[c] * W1[h,c] + b1[h] )
        float s = b1[tid];
#pragma unroll
        for (int c = 0; c < C_; ++c) s = fmaf(ys[c], W1[tid * C_ + c], s);
        h1s[tid] = fmaxf(s, 0.0f);
    }
    __syncthreads();

    if (tid < C_) {   // wave 0 (all 32 lanes active) -> shuffle LN
        float s = b2[tid];
#pragma unroll
        for (int h = 0; h < H_; ++h) s = fmaf(h1s[h], W2[tid * H_ + h], s);
        float z = ys[tid] + s;
        float mean = wred_sum(z) * (1.0f / C_);
        float d = z - mean;
        float var = wred_sum(d * d) * (1.0f / C_) + 1e-14f;
        y2[((size_t)b * C_ + tid) * T_ + t] = d * rsqrtf(var) * gamma2[tid] + beta2[tid];
    }
}

// ---------------- launch ----------------
extern "C" void kernel_launch(void* const* d_in, const int* in_sizes, int n_in,
                              void* d_out, int out_size, void* d_ws, size_t ws_size,
                              hipStream_t stream) {
    const float* x      = (const float*)d_in[0];
    const float* Wt     = (const float*)d_in[1];
    const float* Wx     = (const float*)d_in[2];
    const float* bh     = (const float*)d_in[3];
    const float* Wa     = (const float*)d_in[4];
    const float* ba     = (const float*)d_in[5];
    const float* gamma1 = (const float*)d_in[6];
    const float* beta1  = (const float*)d_in[7];
    const float* W1     = (const float*)d_in[8];
    const float* b1     = (const float*)d_in[9];
    const float* W2     = (const float*)d_in[10];
    const float* b2     = (const float*)d_in[11];
    const float* gamma2 = (const float*)d_in[12];
    const float* beta2  = (const float*)d_in[13];

    float* out   = (float*)d_out;
    float* y2    = out;                              // (B,C,T) = 131072 floats
    float* a_out = out + (size_t)B_ * C_ * T_;       // (B,T,T) = 4194304 floats

    const size_t NBTU = (size_t)B_ * T_ * U_;        // 131072
    const size_t NBCT = (size_t)B_ * C_ * T_;        // 131072
    float* ws = (float*)d_ws;
    float* q  = ws;                                  // (B,T,U)
    float* k  = ws + NBTU;                           // (B,T,U)
    float* v  = ws + 2 * NBTU;                       // (B,C,T)
    float* y  = ws + 2 * NBTU + NBCT;                // (B,C,T)

    qk_kernel<<<(B_ * T_ * U_) / 256, 256, 0, stream>>>(x, Wt, Wx, bh, q, k);
    attn_kernel<<<B_ * T_, 256, 0, stream>>>(q, k, Wa, ba, a_out);
    av_wmma_kernel<<<B_ * (T_ / 16) * (C_ / 16), 32, 0, stream>>>(a_out, x, v);
    ln1_kernel<<<(B_ * T_) / 8, 256, 0, stream>>>(x, v, gamma1, beta1, y);
    ff_ln2_kernel<<<B_ * T_, H_, 0, stream>>>(y, W1, b1, W2, b2, gamma2, beta2, y2);
}